// SSM_56530359550301
// MI455X (gfx1250) — compile-verified
//
#include <hip/hip_runtime.h>

typedef __attribute__((ext_vector_type(2))) float v2f;
typedef __attribute__((ext_vector_type(8))) float v8f;

#define BATCH 4
#define DM 1024
#define SEQ 2048
#define NS 16
#define LC 128                 // chunk length for parallel scan
#define NCHUNK (SEQ / LC)      // 16
#define BC_SCALE 0.25f         // WIDTH_MULT^-0.5 = 16^-0.5
#define RMS_EPS 1.1920929e-07f // finfo(f32).eps

// ---------------------------------------------------------------------------
// Kernel 1: WMMA projections  Bu = s*(B@u), Cu = s*(C@u), stored (B, L, 16).
// Block = 128 threads = 4 waves per (batch, 16-wide L tile); each wave covers
// a K-quarter (256 of 1024) with V_WMMA_F32_16X16X4_F32, partial 16x16 tiles
// reduced through LDS in a fixed order (deterministic). 2048 resident waves.
// ---------------------------------------------------------------------------
__global__ __launch_bounds__(128)
void k_proj_wmma(const float* __restrict__ u, const float* __restrict__ Bm,
                 const float* __restrict__ Cm, float* __restrict__ BuG,
                 float* __restrict__ CuG) {
  __shared__ float redB[4][32][8];
  __shared__ float redC[4][32][8];
  const int b    = blockIdx.x >> 7;           // / 128 tiles
  const int l0   = (blockIdx.x & 127) << 4;   // tile * 16
  const int tid  = threadIdx.x;
  const int lane = tid & 31;
  const int w    = tid >> 5;                  // wave id = K-quarter
  const int half = lane >> 4;                 // K-half select (ISA A/B layout)
  const int lm   = lane & 15;                 // M row (A) / N col (B)

  const float* ub = u + (size_t)b * DM * SEQ;
  const int kbeg = w * (DM / 4);
  const int kend = kbeg + (DM / 4);
  v8f accB = {};
  v8f accC = {};
  for (int k0 = kbeg; k0 < kend; k0 += 4) {
    const int ka = k0 + 2 * half;
    // A-frag: lanes 0-15 hold K=k0,k0+1 of row lm; lanes 16-31 hold K=k0+2,k0+3
    v2f aB = *(const v2f*)(Bm + lm * DM + ka);
    v2f aC = *(const v2f*)(Cm + lm * DM + ka);
    // B-frag: lanes 0-15 hold rows K=k0,k0+1 at col l0+lm; lanes 16-31 rows +2
    v2f bf;
    bf.x = ub[(size_t)ka * SEQ + l0 + lm];
    bf.y = ub[(size_t)(ka + 1) * SEQ + l0 + lm];
    accB = __builtin_amdgcn_wmma_f32_16x16x4_f32(false, aB, false, bf,
                                                 (short)0, accB, false, false);
    accC = __builtin_amdgcn_wmma_f32_16x16x4_f32(false, aC, false, bf,
                                                 (short)0, accC, false, false);
  }
#pragma unroll
  for (int r = 0; r < 8; ++r) {
    redB[w][lane][r] = accB[r];
    redC[w][lane][r] = accC[r];
  }
  __syncthreads();
  if (w == 0) {
    // D layout: VGPR r -> row (r + 8*half), col lm. Store as (b, l, n).
    const int l = l0 + lm;
#pragma unroll
    for (int r = 0; r < 8; ++r) {
      const int n = r + 8 * half;
      const float sb = (redB[0][lane][r] + redB[1][lane][r]) +
                       (redB[2][lane][r] + redB[3][lane][r]);
      const float sc = (redC[0][lane][r] + redC[1][lane][r]) +
                       (redC[2][lane][r] + redC[3][lane][r]);
      BuG[((size_t)(b * SEQ + l)) * NS + n] = sb * BC_SCALE;
      CuG[((size_t)(b * SEQ + l)) * NS + n] = sc * BC_SCALE;
    }
  }
}

// ---------------------------------------------------------------------------
// Kernel 2: chunked local scan. Block = (b, 256-wide d chunk, 128-long l
// chunk). u staged via padded LDS tile: b128 global loads (coalesced),
// scalar LDS stores keep the 33-float pad so per-thread column reads are
// conflict-free. Next subtile prefetched (global_prefetch_b8) during compute.
// Writes tmp in (B, L, D) layout (coalesced) and the chunk-final state h_end.
// ---------------------------------------------------------------------------
__global__ __launch_bounds__(256)
void k_chunk_scan(const float* __restrict__ u, const float* __restrict__ A,
                  const float* __restrict__ Dv, const float* __restrict__ BuG,
                  const float* __restrict__ CuG, float* __restrict__ tmp,
                  float* __restrict__ hEnd) {
  __shared__ float su[256][33];   // 256 d x 32 l, padded
  __shared__ float sbu[32][NS];
  __shared__ float scu[32][NS];

  const int gid = blockIdx.x;
  const int b  = gid >> 6;
  const int dc = (gid >> 4) & 3;
  const int c  = gid & 15;
  const int d0 = dc << 8;
  const int l0 = c * LC;
  const int tid = threadIdx.x;
  const int d = d0 + tid;

  float a_[NS];
#pragma unroll
  for (int n = 0; n < NS; ++n) a_[n] = A[n];
  const float dd = Dv[d];
  float h[NS];
#pragma unroll
  for (int n = 0; n < NS; ++n) h[n] = 0.f;

  const float* ub = u + (size_t)b * DM * SEQ;

  for (int s = 0; s < LC / 32; ++s) {
    const int lb = l0 + s * 32;
    {   // stage u tile: 8 b128 loads per thread; 128B segments per 8 lanes
      const int c4 = (tid & 7) << 2;  // 0..28 step 4
      const int r0 = tid >> 3;        // 0..31
#pragma unroll
      for (int p = 0; p < 8; ++p) {
        const int row = r0 + 32 * p;
        const float4 v = *(const float4*)(ub + (size_t)(d0 + row) * SEQ + lb + c4);
        su[row][c4 + 0] = v.x;
        su[row][c4 + 1] = v.y;
        su[row][c4 + 2] = v.z;
        su[row][c4 + 3] = v.w;
      }
    }
    {   // stage Bu/Cu (contiguous 2KB each)
#pragma unroll
      for (int i = 0; i < 2; ++i) {
        const int idx = tid + 256 * i;
        const int t = idx >> 4;
        const int n = idx & 15;
        sbu[t][n] = BuG[((size_t)(b * SEQ + lb + t)) * NS + n];
        scu[t][n] = CuG[((size_t)(b * SEQ + lb + t)) * NS + n];
      }
    }
    // prefetch next 32-l subtile: one byte per 128B-aligned row segment
    if (s + 1 < LC / 32) {
      __builtin_prefetch(ub + (size_t)(d0 + tid) * SEQ + lb + 32, 0, 3);
    }
    __syncthreads();
    for (int t = 0; t < 32; ++t) {
      const float uv = su[tid][t];
      float y = 0.f;
#pragma unroll
      for (int n = 0; n < NS; ++n) {
        h[n] = a_[n] * h[n] + sbu[t][n] * uv;
        y += scu[t][n] * h[n];
      }
      tmp[((size_t)(b * SEQ + lb + t)) * DM + d] = y + uv * dd;
    }
    __syncthreads();
  }
  const size_t base = ((size_t)((b * NCHUNK + c) * DM) + d) * NS;
#pragma unroll
  for (int n = 0; n < NS; ++n) hEnd[base + n] = h[n];
}

// ---------------------------------------------------------------------------
// Kernel 3: combine chunk states. h_start[c] = A^LC * h_start[c-1] + hEnd[c-1]
// One thread per (b,d). A^128 by 7 squarings (exact).
// ---------------------------------------------------------------------------
__global__ __launch_bounds__(256)
void k_chunk_combine(const float* __restrict__ A, const float* __restrict__ hEnd,
                     float* __restrict__ hStart) {
  const int idx = blockIdx.x * 256 + threadIdx.x; // b*DM + d
  const int b = idx >> 10;
  const int d = idx & 1023;
  float aP[NS];
#pragma unroll
  for (int n = 0; n < NS; ++n) {
    float x = A[n];
    for (int i = 0; i < 7; ++i) x = x * x;  // A^128
    aP[n] = x;
  }
  float h[NS];
#pragma unroll
  for (int n = 0; n < NS; ++n) h[n] = 0.f;
  for (int c = 0; c < NCHUNK; ++c) {
    const size_t base = ((size_t)((b * NCHUNK + c) * DM) + d) * NS;
#pragma unroll
    for (int n = 0; n < NS; ++n) {
      hStart[base + n] = h[n];
      h[n] = aP[n] * h[n] + hEnd[base + n];
    }
  }
}

// ---------------------------------------------------------------------------
// Kernel 4: cross-chunk correction fused with RMS sum-of-squares.
// One block per (b,l):  y += sum_n Cu[n,l] * A_n^{t+1} * h_start[c][n]
// then block-reduce sum(v*v) over d -> sums[b,l]. Fully coalesced, no atomics.
// ---------------------------------------------------------------------------
__global__ __launch_bounds__(256)
void k_correct_reduce(const float* __restrict__ A, const float* __restrict__ CuG,
                      const float* __restrict__ hStart, float* __restrict__ tmp,
                      float* __restrict__ sums) {
  __shared__ float w[NS];
  __shared__ float red[256];
  const int bl = blockIdx.x;       // b*SEQ + l
  const int b = bl >> 11;
  const int l = bl & (SEQ - 1);
  const int c = l / LC;
  const int t = l % LC;
  const int tid = threadIdx.x;
  if (tid < NS) {
    float p = 1.f;
    const float a = A[tid];
    for (int i = 0; i <= t; ++i) p *= a;          // A^(t+1)
    w[tid] = CuG[(size_t)bl * NS + tid] * p;
  }
  __syncthreads();
  float acc = 0.f;
#pragma unroll
  for (int i = 0; i < DM / 256; ++i) {
    const int d = tid + 256 * i;
    const size_t ti = (size_t)bl * DM + d;
    float v = tmp[ti];
    const size_t base = ((size_t)((b * NCHUNK + c) * DM) + d) * NS;
#pragma unroll
    for (int n = 0; n < NS; ++n) v += w[n] * hStart[base + n];
    tmp[ti] = v;
    acc += v * v;
  }
  red[tid] = acc;
  __syncthreads();
  for (int s = 128; s > 0; s >>= 1) {
    if (tid < s) red[tid] += red[tid + s];
    __syncthreads();
  }
  if (tid == 0) sums[bl] = red[0];
}

// ---------------------------------------------------------------------------
// Kernel 5: RMS-normalize + transpose (B,L,D) -> (B,D,L) via 32x33 LDS tile.
// Both the read and the write are coalesced.
// ---------------------------------------------------------------------------
__global__ __launch_bounds__(256)
void k_norm_transpose(const float* __restrict__ tmp, const float* __restrict__ sums,
                      const float* __restrict__ wrms, float* __restrict__ out) {
  __shared__ float tile[32][33];
  const int l0 = blockIdx.x << 5;
  const int d0 = blockIdx.y << 5;
  const int b  = blockIdx.z;
  const int tx = threadIdx.x;
  const int ty = threadIdx.y;
#pragma unroll
  for (int i = 0; i < 4; ++i) {
    const int r = ty + 8 * i;   // l offset
    tile[r][tx] = tmp[((size_t)(b * SEQ + l0 + r)) * DM + d0 + tx];
  }
  __syncthreads();
#pragma unroll
  for (int i = 0; i < 4; ++i) {
    const int dy = ty + 8 * i;  // d offset
    const float s = sums[b * SEQ + l0 + tx];
    const float inv = rsqrtf(s * (1.0f / DM) + RMS_EPS);
    out[((size_t)(b * DM + d0 + dy)) * SEQ + l0 + tx] =
        tile[tx][dy] * inv * wrms[d0 + dy];
  }
}

// ---------------------------------------------------------------------------
extern "C" void kernel_launch(void* const* d_in, const int* in_sizes, int n_in,
                              void* d_out, int out_size, void* d_ws, size_t ws_size,
                              hipStream_t stream) {
  const float* u    = (const float*)d_in[0];
  const float* A    = (const float*)d_in[1];
  const float* Bm   = (const float*)d_in[2];
  const float* Cm   = (const float*)d_in[3];
  const float* Dv   = (const float*)d_in[4];
  const float* wrms = (const float*)d_in[5];
  // d_in[6] = L (2048), hard-coded above.

  float* ws = (float*)d_ws;
  const size_t nBu  = (size_t)BATCH * SEQ * NS;         // 131072
  const size_t nTmp = (size_t)BATCH * SEQ * DM;         // 8388608
  const size_t nH   = (size_t)BATCH * NCHUNK * DM * NS; // 1048576
  float* BuG    = ws;
  float* CuG    = BuG + nBu;
  float* tmp    = CuG + nBu;
  float* hEnd   = tmp + nTmp;
  float* hStart = hEnd + nH;
  float* sums   = hStart + nH;   // + BATCH*SEQ floats

  k_proj_wmma<<<BATCH * (SEQ / 16), 128, 0, stream>>>(u, Bm, Cm, BuG, CuG);
  k_chunk_scan<<<BATCH * 4 * NCHUNK, 256, 0, stream>>>(u, A, Dv, BuG, CuG, tmp, hEnd);
  k_chunk_combine<<<(BATCH * DM) / 256, 256, 0, stream>>>(A, hEnd, hStart);
  k_correct_reduce<<<BATCH * SEQ, 256, 0, stream>>>(A, CuG, hStart, tmp, sums);
  dim3 g3(SEQ / 32, DM / 32, BATCH);
  k_norm_transpose<<<g3, dim3(32, 8), 0, stream>>>(tmp, sums, wrms, (float*)d_out);
}